// CrossEntropyLoss_81741817577604
// MI455X (gfx1250) — compile-verified
//
#include <hip/hip_runtime.h>
#include <math.h>

// native clang vector type: valid for __builtin_nontemporal_load
typedef float f32x4 __attribute__((ext_vector_type(4)));

// ---------------------------------------------------------------------------
// CDNA5 async global->LDS copy (tracked by ASYNCcnt) + wait. These are the
// gfx1250-specific data-movement instructions (see CDNA5 ISA).
// Streamed-once data: mark non-temporal (TH_LOAD_NT) to avoid L2 pollution
// (524 MB stream vs 192 MB L2).
// ---------------------------------------------------------------------------
#define WAIT_ASYNCCNT(n) asm volatile("s_wait_asynccnt " #n ::: "memory")

__device__ __forceinline__ void async_load_b128_nt(unsigned lds_byte_off, const void* gaddr) {
  // LDS dest = LDS base + offset VGPR + imm offset ; global addr = 64-bit VGPR pair
  asm volatile("global_load_async_to_lds_b128 %0, %1, off th:TH_LOAD_NT"
               :: "v"(lds_byte_off), "v"(gaddr)
               : "memory");
}

// merge running (m, s) with (m2, s2) in log-sum-exp form
__device__ __forceinline__ void online_merge(float& m, float& s, float m2, float s2) {
  float M = fmaxf(m, m2);
  s = s * __expf(m - M) + s2 * __expf(m2 - M);
  m = M;
}

// ---------------------------------------------------------------------------
// One block (256 threads = 8 wave32) per row. Single pass over the row using
// online softmax; row logits streamed through LDS via async b128 loads
// (double-buffered 2 x 16KB), tail handled with direct float4 loads.
// Each thread reads back only the bytes it async-loaded itself, so after
// s_wait_asynccnt no cross-wave barrier is needed in the steady-state loop.
// ---------------------------------------------------------------------------
__global__ __launch_bounds__(256) void ce_row_kernel(const float* __restrict__ logits,
                                                     const int* __restrict__ labels,
                                                     float* __restrict__ row_loss,
                                                     int V) {
  __shared__ __align__(16) float sh[8192];   // 2 buffers x 4096 floats (16KB each)
  __shared__ float2 wred[8];

  const int tid = threadIdx.x;
  const int row = blockIdx.x;
  const float* rowp = logits + (size_t)row * (size_t)V;

  // Launder the LDS pointer/offset: the buffer is only written by the async
  // DMA (side effect invisible to the compiler), so force the compiler to
  // treat its contents as unknown.
  float* shp = sh;
  unsigned sbase = (unsigned)(uintptr_t)sh;   // addr[31:0] of LDS flat addr == DS offset
  asm volatile("" : "+v"(shp), "+v"(sbase) :: "memory");

  const int V4    = V >> 2;       // # of float4 elements (V % 4 == 0 here)
  const int nfull = V4 >> 10;     // full 1024-float4 (16KB) chunks

  float m = -3.402823466e38f;     // running max (finite sentinel: avoids inf-inf NaN)
  float s = 0.0f;                 // running sum of exp(x - m)

  auto issue = [&](int c) {
    const unsigned bufb = (c & 1) ? 16384u : 0u;
    const char* g = (const char*)rowp + (size_t)c * 16384u;
#pragma unroll
    for (int k = 0; k < 4; ++k) {
      const unsigned slot = (unsigned)(k * 256 + tid) * 16u;  // same layout in LDS & mem
      async_load_b128_nt(sbase + bufb + slot, g + slot);
    }
  };

  if (nfull > 0) issue(0);
  for (int c = 0; c < nfull; ++c) {
    if (c + 1 < nfull) { issue(c + 1); WAIT_ASYNCCNT(4); }  // chunk c done (in-order)
    else               { WAIT_ASYNCCNT(0); }
    const float* buf = shp + ((c & 1) ? 4096 : 0);

    f32x4 a0 = *(const f32x4*)(buf + (0 * 256 + tid) * 4);
    f32x4 a1 = *(const f32x4*)(buf + (1 * 256 + tid) * 4);
    f32x4 a2 = *(const f32x4*)(buf + (2 * 256 + tid) * 4);
    f32x4 a3 = *(const f32x4*)(buf + (3 * 256 + tid) * 4);

    float mc = fmaxf(fmaxf(fmaxf(fmaxf(a0.x, a0.y), fmaxf(a0.z, a0.w)),
                           fmaxf(fmaxf(a1.x, a1.y), fmaxf(a1.z, a1.w))),
                     fmaxf(fmaxf(fmaxf(a2.x, a2.y), fmaxf(a2.z, a2.w)),
                           fmaxf(fmaxf(a3.x, a3.y), fmaxf(a3.z, a3.w))));
    float M = fmaxf(m, mc);
    float t = s * __expf(m - M);
    t += __expf(a0.x - M) + __expf(a0.y - M) + __expf(a0.z - M) + __expf(a0.w - M);
    t += __expf(a1.x - M) + __expf(a1.y - M) + __expf(a1.z - M) + __expf(a1.w - M);
    t += __expf(a2.x - M) + __expf(a2.y - M) + __expf(a2.z - M) + __expf(a2.w - M);
    t += __expf(a3.x - M) + __expf(a3.y - M) + __expf(a3.z - M) + __expf(a3.w - M);
    s = t;
    m = M;
  }

  // Tail float4s (for V=32000: indices 7168..7999 -> 3328 floats), direct NT loads.
  for (int j = (nfull << 10) + tid; j < V4; j += 256) {
    const f32x4* p4 = (const f32x4*)(rowp + (size_t)j * 4);
    f32x4 a = __builtin_nontemporal_load(p4);
    float mc = fmaxf(fmaxf(a.x, a.y), fmaxf(a.z, a.w));
    float M = fmaxf(m, mc);
    s = s * __expf(m - M)
      + __expf(a.x - M) + __expf(a.y - M) + __expf(a.z - M) + __expf(a.w - M);
    m = M;
  }

  // wave32 butterfly reduce of (m, s)
#pragma unroll
  for (int o = 16; o > 0; o >>= 1) {
    float mo = __shfl_xor(m, o, 32);
    float so = __shfl_xor(s, o, 32);
    online_merge(m, s, mo, so);
  }

  const int wave = tid >> 5;
  if ((tid & 31) == 0) wred[wave] = make_float2(m, s);
  __syncthreads();

  if (tid == 0) {
    float M = wred[0].x, S = wred[0].y;
#pragma unroll
    for (int w = 1; w < 8; ++w) online_merge(M, S, wred[w].x, wred[w].y);
    const int lbl = labels[row];
    const float lt = rowp[lbl];
    const float p = expf(lt - M) / S;          // softmax prob of target class
    row_loss[row] = -logf(p + 1e-12f);         // matches reference EPS
  }
}

// ---------------------------------------------------------------------------
// Deterministic fixed-order reduction of per-row losses -> mean. One block.
// ---------------------------------------------------------------------------
__global__ __launch_bounds__(256) void finalize_kernel(const float* __restrict__ row_loss,
                                                       float* __restrict__ out, int B) {
  __shared__ float red[256];
  const int tid = threadIdx.x;
  float acc = 0.0f;
  for (int i = tid; i < B; i += 256) acc += row_loss[i];  // fixed per-thread order
  red[tid] = acc;
  __syncthreads();
#pragma unroll
  for (int stride = 128; stride > 0; stride >>= 1) {
    if (tid < stride) red[tid] += red[tid + stride];
    __syncthreads();
  }
  if (tid == 0) out[0] = red[0] / (float)B;
}

// ---------------------------------------------------------------------------
extern "C" void kernel_launch(void* const* d_in, const int* in_sizes, int n_in,
                              void* d_out, int out_size, void* d_ws, size_t ws_size,
                              hipStream_t stream) {
  const float* logits = (const float*)d_in[0];
  const int*   labels = (const int*)d_in[1];
  const int B = in_sizes[1];
  const int V = in_sizes[0] / B;

  float* row_loss = (float*)d_ws;  // B floats of scratch

  ce_row_kernel<<<B, 256, 0, stream>>>(logits, labels, row_loss, V);
  finalize_kernel<<<1, 256, 0, stream>>>(row_loss, (float*)d_out, B);
}